// MLLAttentionWithGroupedNorm_70093866270966
// MI455X (gfx1250) — compile-verified
//
#include <hip/hip_runtime.h>
#include <math.h>

typedef float v2f __attribute__((ext_vector_type(2)));
typedef float v8f __attribute__((ext_vector_type(8)));
typedef unsigned int v4u __attribute__((ext_vector_type(4)));
typedef int v4i __attribute__((ext_vector_type(4)));
typedef int v8i __attribute__((ext_vector_type(8)));

#define B_    8
#define C_    512
#define H_    96
#define W_    96
#define N_    9216      // 96*96
#define HEADS_ 4
#define HD_   128
#define KMAX_ 256       // c/2 rope pairs

// ---------------------------------------------------------------------------
// RoPE cos/sin table: angle(h_i, kp) = h_i * 10000^(-kp/256)
// ---------------------------------------------------------------------------
__global__ __launch_bounds__(256) void mlla_rope_table(float2* __restrict__ tab) {
    int idx = blockIdx.x * 256 + threadIdx.x;        // 96*256 entries
    int hi = idx >> 8;
    int kp = idx & 255;
    float theta = __expf(-(float)kp * (logf(10000.0f) / 256.0f));
    float ang = (float)hi * theta;
    float s, c;
    __sincosf(ang, &s, &c);
    tab[idx] = make_float2(c, s);
}

// ---------------------------------------------------------------------------
// qk = xs @ Wqk^T + bias, then elu+1. xs[m,ch] = x[b,ch,m].
// Block: 8 waves; wave = 16 m-rows x 128 j-cols (8 WMMA tiles sharing A).
// grid (72 m-blocks, 8 j-blocks of 128, B). j-block<4 -> q, else -> k.
// ---------------------------------------------------------------------------
__global__ __launch_bounds__(256)
void mlla_qk_gemm(const float* __restrict__ x, const float* __restrict__ Wqk,
                  const float* __restrict__ bqk,
                  float* __restrict__ q, float* __restrict__ k) {
    const int b    = blockIdx.z;
    const int jb   = blockIdx.y;
    const int mb   = blockIdx.x;
    const int wave = threadIdx.x >> 5;
    const int lane = threadIdx.x & 31;
    const int l    = lane & 15;
    const int lhi  = lane >> 4;
    const int kkb0 = lhi ? 2 : 0;
    const int mrow = mb * 128 + wave * 16 + l;

    const float* xb = x + (size_t)b * C_ * N_;

    v8f acc[8];
    for (int t = 0; t < 8; ++t)
        for (int i = 0; i < 8; ++i) acc[t][i] = 0.0f;

    for (int kk0 = 0; kk0 < C_; kk0 += 4) {
        v2f a;
        a.x = xb[(size_t)(kk0 + kkb0    ) * N_ + mrow];
        a.y = xb[(size_t)(kk0 + kkb0 + 1) * N_ + mrow];
#pragma unroll
        for (int t = 0; t < 8; ++t) {
            int j = jb * 128 + t * 16 + l;
            float2 wv = *(const float2*)&Wqk[(size_t)j * C_ + kk0 + kkb0];
            v2f bm; bm.x = wv.x; bm.y = wv.y;
            acc[t] = __builtin_amdgcn_wmma_f32_16x16x4_f32(
                false, a, false, bm, (short)0, acc[t], false, false);
        }
    }

    // epilogue: bias + elu+1, store to q (jb<4) or k (jb>=4)
#pragma unroll
    for (int t = 0; t < 8; ++t) {
        int j = jb * 128 + t * 16 + l;
        float bias = bqk[j];
        float* dst = (jb < 4) ? (q + (size_t)b * N_ * C_ + j)
                              : (k + (size_t)b * N_ * C_ + (j - C_));
#pragma unroll
        for (int r = 0; r < 8; ++r) {
            int m = mb * 128 + wave * 16 + r + lhi * 8;
            float v = acc[t][r] + bias;
            v = (v > 0.0f) ? (v + 1.0f) : __expf(v);
            dst[(size_t)m * C_] = v;
        }
    }
}

// ---------------------------------------------------------------------------
// k_mean[b,head,d] = (1/n) sum over quarter-range rows & 4 channel blocks
// ---------------------------------------------------------------------------
__global__ __launch_bounds__(256)
void mlla_kmean(const float* __restrict__ k, float* __restrict__ kmean) {
    int bh = blockIdx.x;
    int b = bh >> 2, head = bh & 3;
    int d = threadIdx.x & 127;
    int half = threadIdx.x >> 7;
    const float* kb = k + (size_t)b * N_ * C_;
    int p0 = head * 2304 + half * 1152;
    float s = 0.0f;
    for (int p = p0; p < p0 + 1152; ++p) {
        const float* row = kb + (size_t)p * C_ + d;
        s += row[0] + row[128] + row[256] + row[384];
    }
    __shared__ float red[256];
    red[threadIdx.x] = s;
    __syncthreads();
    if (threadIdx.x < 128)
        kmean[bh * HD_ + d] = (red[threadIdx.x] + red[threadIdx.x + 128]) * (1.0f / (float)N_);
}

// ---------------------------------------------------------------------------
// z[b,head,m] = 1/(dot(q2[b,head,m,:], k_mean) + 1e-6)
// q2 via flat reshape: g=head*n+m -> row g>>2, channels (g&3)*128..+127
// ---------------------------------------------------------------------------
__global__ __launch_bounds__(256)
void mlla_z(const float* __restrict__ q, const float* __restrict__ kmean,
            float* __restrict__ z) {
    int bh = blockIdx.y;
    int b = bh >> 2, head = bh & 3;
    int m = blockIdx.x * 256 + threadIdx.x;
    int g = head * N_ + m;
    const float4* qp = (const float4*)(q + ((size_t)b * N_ + (g >> 2)) * C_ + (size_t)(g & 3) * HD_);
    const float4* kp = (const float4*)(kmean + (size_t)bh * HD_);
    float s = 0.0f;
#pragma unroll 8
    for (int i = 0; i < HD_ / 4; ++i) {
        float4 a = qp[i], c = kp[i];
        s += a.x * c.x + a.y * c.y + a.z * c.z + a.w * c.w;
    }
    z[(size_t)bh * N_ + m] = 1.0f / (s + 1e-6f);
}

// ---------------------------------------------------------------------------
// kv[b,head,d,e] = (1/n) * sum_m k_rope[m,d] * v2[m,e]
// A = k_rope (RoPE applied on load from table), B = v2 via flat-reshape map.
// Split-K over m (16 splits of 576), accumulate with f32 atomics.
// ---------------------------------------------------------------------------
__global__ __launch_bounds__(256)
void mlla_kv_gemm(const float* __restrict__ x, const float* __restrict__ k,
                  const float2* __restrict__ tab, float* __restrict__ kv) {
    int bh = blockIdx.y;
    int b = bh >> 2, head = bh & 3;
    int split = blockIdx.x;
    int wave = threadIdx.x >> 5, lane = threadIdx.x & 31;
    int l = lane & 15, lhi = lane >> 4;
    int dstrip = wave * 16;
    int kkb0 = lhi ? 2 : 0;

    int ch  = head * HD_ + dstrip + l;    // per-lane A channel
    int chE = ch & ~1;
    int odd = ch & 1;
    int kp  = ch >> 1;

    const float* kb = k + (size_t)b * N_ * C_;
    const float* xb = x + (size_t)b * C_ * N_;

    v8f acc[8];
    for (int t = 0; t < 8; ++t)
        for (int i = 0; i < 8; ++i) acc[t][i] = 0.0f;

    int m0 = split * 576;
    for (int mm = m0; mm < m0 + 576; mm += 4) {
        v2f a;
#pragma unroll
        for (int kk = 0; kk < 2; ++kk) {
            int m = mm + kkb0 + kk;
            float2 pr = *(const float2*)&kb[(size_t)m * C_ + chE];
            float2 cs = tab[(m / 96) * KMAX_ + kp];
            float val = odd ? (cs.y * pr.x + cs.x * pr.y)
                            : (cs.x * pr.x - cs.y * pr.y);
            if (kk == 0) a.x = val; else a.y = val;
        }
#pragma unroll
        for (int t = 0; t < 8; ++t) {
            int e = t * 16 + l;
            v2f bm;
#pragma unroll
            for (int kk = 0; kk < 2; ++kk) {
                int m = mm + kkb0 + kk;
                int g = head * N_ + m;
                float v = xb[(size_t)((g & 3) * HD_ + e) * N_ + (g >> 2)];
                if (kk == 0) bm.x = v; else bm.y = v;
            }
            acc[t] = __builtin_amdgcn_wmma_f32_16x16x4_f32(
                false, a, false, bm, (short)0, acc[t], false, false);
        }
    }

    const float sc = 1.0f / (float)N_;   // scale^2 = (n^-0.5)^2
    float* kvb = kv + (size_t)bh * HD_ * HD_;
#pragma unroll
    for (int t = 0; t < 8; ++t)
#pragma unroll
        for (int r = 0; r < 8; ++r) {
            int d = dstrip + r + lhi * 8;
            int e = t * 16 + l;
            atomicAdd(&kvb[d * HD_ + e], acc[t][r] * sc);
        }
}

// ---------------------------------------------------------------------------
// out^T[e,m] = sum_d kv[d,e] * q_rope[m,d];  val = out*z + depthwise3x3(x)+bias
// kv tile (64KB, contiguous) is DMA'd into LDS by the Tensor Data Mover:
// wave 0 issues tensor_load_to_lds with a D# descriptor, waits TENSORcnt==0,
// then the workgroup barrier releases the other waves. kvs is the only
// __shared__ object -> LDS offset 0 in the descriptor.
// Transposed GEMM so the big 151MB store is lane-contiguous along m.
// ---------------------------------------------------------------------------
__global__ __launch_bounds__(256)
void mlla_out(const float* __restrict__ q, const float* __restrict__ kv,
              const float* __restrict__ z, const float2* __restrict__ tab,
              const float* __restrict__ x, const float* __restrict__ lw,
              const float* __restrict__ lb, float* __restrict__ out) {
    __shared__ float kvs[HD_ * HD_];
    int bh = blockIdx.y;
    int b = bh >> 2, head = bh & 3;
    int mb = blockIdx.x;

    if (threadIdx.x < 32) {  // wave 0 issues the TDM load (EXEC-independent)
        uint64_t ga = (uint64_t)(uintptr_t)(kv + (size_t)bh * HD_ * HD_);
        v4u g0;
        g0.x = 1u;                                   // count=1 (valid D#)
        g0.y = 0u;                                   // lds_addr = offset of kvs
        g0.z = (unsigned)(ga & 0xFFFFFFFFu);         // global_addr[31:0]
        g0.w = (unsigned)((ga >> 32) & 0x1FFFFFFu)   // global_addr[56:32]
             | (2u << 30);                           // type = 2 (image/tensor)
        v8i g1;
        g1[0] = 0x20000;                             // data_size=2 -> 4 bytes
        g1[1] = (int)(16384u << 16);                 // tensor_dim0[15:0]=16384
        g1[2] = (int)(1u << 16);                     // dim0 hi=0; tensor_dim1=1
        g1[3] = (int)(16384u << 16);                 // tile_dim0 = 16384
        g1[4] = 1;                                   // tile_dim1 = 1
        g1[5] = 16384;                               // tensor_dim0_stride lo
        g1[6] = 0;
        g1[7] = 0;
        v4i g2 = {0, 0, 0, 0};                       // <=2D: groups 2/3 unused
        v4i g3 = {0, 0, 0, 0};
        v8i g4 = {0, 0, 0, 0, 0, 0, 0, 0};           // trailing group (6-arg form)
        __builtin_amdgcn_tensor_load_to_lds(g0, g1, g2, g3, g4, 0);
        __builtin_amdgcn_s_wait_tensorcnt(0);
    }
    __syncthreads();

    int wave = threadIdx.x >> 5, lane = threadIdx.x & 31;
    int l = lane & 15, lhi = lane >> 4;
    int estrip = wave * 16;
    int kkb0 = lhi ? 2 : 0;
    const float* qb = q + (size_t)b * N_ * C_;

    v8f acc[8];
    for (int t = 0; t < 8; ++t)
        for (int i = 0; i < 8; ++i) acc[t][i] = 0.0f;

    for (int d0 = 0; d0 < HD_; d0 += 4) {
        v2f a;
        a.x = kvs[(d0 + kkb0    ) * HD_ + estrip + l];
        a.y = kvs[(d0 + kkb0 + 1) * HD_ + estrip + l];
        int chq = head * HD_ + d0 + kkb0;   // even channel
        int kp = chq >> 1;
#pragma unroll
        for (int t = 0; t < 8; ++t) {
            int m = mb * 128 + t * 16 + l;
            float2 qv = *(const float2*)&qb[(size_t)m * C_ + chq];
            float2 cs = tab[(m / 96) * KMAX_ + kp];
            v2f bm;
            bm.x = cs.x * qv.x - cs.y * qv.y;   // rotated even channel
            bm.y = cs.y * qv.x + cs.x * qv.y;   // rotated odd channel
            acc[t] = __builtin_amdgcn_wmma_f32_16x16x4_f32(
                false, a, false, bm, (short)0, acc[t], false, false);
        }
    }

    // epilogue: * z, + depthwise 3x3 conv (lepe) + bias, coalesced store
    const float* zb = z + (size_t)bh * N_;
    float zarr[8];
#pragma unroll
    for (int t = 0; t < 8; ++t) zarr[t] = zb[mb * 128 + t * 16 + l];

#pragma unroll
    for (int r = 0; r < 8; ++r) {
        int e = estrip + r + lhi * 8;
        int ch = head * HD_ + e;
        float wgt[9];
#pragma unroll
        for (int i = 0; i < 9; ++i) wgt[i] = lw[ch * 9 + i];
        float bias = lb[ch];
        const float* xc = x + ((size_t)b * C_ + ch) * N_;
        float* oc = out + ((size_t)b * C_ + ch) * N_;
#pragma unroll
        for (int t = 0; t < 8; ++t) {
            int m = mb * 128 + t * 16 + l;
            int hy = m / 96;
            int wj = m - hy * 96;
            float cv = bias;
#pragma unroll
            for (int dy = -1; dy <= 1; ++dy) {
                int yy = hy + dy;
                if ((unsigned)yy < 96u) {
                    const float* row = xc + yy * 96;
#pragma unroll
                    for (int dx = -1; dx <= 1; ++dx) {
                        int xx = wj + dx;
                        if ((unsigned)xx < 96u)
                            cv += row[xx] * wgt[(dy + 1) * 3 + (dx + 1)];
                    }
                }
            }
            oc[m] = acc[t][r] * zarr[t] + cv;
        }
    }
}

// ---------------------------------------------------------------------------
extern "C" void kernel_launch(void* const* d_in, const int* in_sizes, int n_in,
                              void* d_out, int out_size, void* d_ws, size_t ws_size,
                              hipStream_t stream) {
    (void)in_sizes; (void)n_in; (void)out_size; (void)ws_size;
    const float* x   = (const float*)d_in[0];
    const float* Wqk = (const float*)d_in[1];
    const float* bqk = (const float*)d_in[2];
    const float* lw  = (const float*)d_in[3];
    const float* lb  = (const float*)d_in[4];
    float* out = (float*)d_out;

    float* ws = (float*)d_ws;
    const size_t QN = (size_t)B_ * N_ * C_;              // 37.7M floats
    float*  q     = ws;
    float*  k     = q + QN;
    float*  kv    = k + QN;                              // 8*4*128*128
    float*  kmean = kv + (size_t)B_ * HEADS_ * HD_ * HD_;
    float*  z     = kmean + (size_t)B_ * HEADS_ * HD_;
    float2* tab   = (float2*)(z + (size_t)B_ * HEADS_ * N_);  // 96*256 float2

    (void)hipMemsetAsync(kv, 0, sizeof(float) * B_ * HEADS_ * HD_ * HD_, stream);
    mlla_rope_table<<<96, 256, 0, stream>>>(tab);
    mlla_qk_gemm<<<dim3(72, 8, B_), 256, 0, stream>>>(x, Wqk, bqk, q, k);
    mlla_kmean<<<B_ * HEADS_, 256, 0, stream>>>(k, kmean);
    mlla_z<<<dim3(36, B_ * HEADS_), 256, 0, stream>>>(q, kmean, z);
    mlla_kv_gemm<<<dim3(16, B_ * HEADS_), 256, 0, stream>>>(x, k, tab, kv);
    mlla_out<<<dim3(72, B_ * HEADS_), 256, 0, stream>>>(q, kv, z, tab, x, lw, lb, out);
}